// WorkingMemory_88304527606267
// MI455X (gfx1250) — compile-verified
//
#include <hip/hip_runtime.h>

typedef __attribute__((ext_vector_type(2))) float v2f;
typedef __attribute__((ext_vector_type(8))) float v8f;

#define TT 128
#define NB 2048
#define NSLOT 7
#define SS 50
#define KP 52                         // K padded to multiple of 4 (13 wmma k-steps)
#define NPAD 64                       // N padded to 4 tiles of 16
#define ROW_STRIDE (NSLOT * SS)       // 350
#define STEP_STRIDE (NB * ROW_STRIDE) // 716800 floats per timestep

// One block: 4 waves, each wave owns a 16-row batch tile of one slot.
// Grid: 7 slots * (2048/64) batch-blocks = 224 blocks.
__global__ __launch_bounds__(128)
void wm_lif_kernel(const float* __restrict__ pattern,
                   const float* __restrict__ noise,
                   const float* __restrict__ W,
                   float* __restrict__ out)
{
    // Wm TRANSPOSED: WldsT[n][k] = Wm[k][n] * 0.99, diag masked, zero padded.
    // B fragment {Wm[k][n], Wm[k+1][n]} is then CONTIGUOUS -> one ds_load_b64
    // directly into an even-aligned VGPR pair (no regrouping moves).
    __shared__ __align__(16) float WldsT[NPAD * KP];
    // Per-wave spike tile [M=16][K=52]; cols 50,51 stay zero forever.
    __shared__ __align__(16) float Spk[4][16 * KP];

    const int tid  = threadIdx.x;
    const int lane = tid & 31;
    const int wave = tid >> 5;
    const int hi   = lane >> 4;       // lane half
    const int nlo  = lane & 15;
    const int koff = hi * 2;          // K offset for A/B fragments in upper half-wave

    const int slot    = blockIdx.x / (NB / 64);
    const int bblk    = blockIdx.x % (NB / 64);
    const int rowBase = bblk * 64 + wave * 16;

    // ---- init LDS: zero, then fill transposed masked/scaled W ----
    for (int i = tid; i < NPAD * KP; i += 128) WldsT[i] = 0.0f;
    for (int i = tid; i < 16 * KP; i += 128) {
        Spk[0][i] = 0.0f; Spk[1][i] = 0.0f; Spk[2][i] = 0.0f; Spk[3][i] = 0.0f;
    }
    __syncthreads();
    for (int i = tid; i < SS * SS; i += 128) {
        int k = i / SS, n = i % SS;
        float w = W[(slot * SS + k) * SS + n];
        WldsT[n * KP + k] = (k == n) ? 0.0f : w * 0.99f;
    }
    __syncthreads();

    const float alpha = expf(-1.0f / 30.0f);  // LIF decay

    // v state in registers, laid out exactly like the WMMA C/D fragment:
    // element [ntile][r] <-> (M = r + 8*hi, N = ntile*16 + nlo)
    float vst[4][8];
#pragma unroll
    for (int q = 0; q < 4; ++q)
#pragma unroll
        for (int r = 0; r < 8; ++r) vst[q][r] = 0.0f;

    // Per-lane global flat base: row (rowBase + 8*hi), col slot*50 + nlo.
    const int baseFlat = (rowBase + 8 * hi) * ROW_STRIDE + slot * SS + nlo;
    const bool valid3  = (nlo < 2);   // N-tile 3 covers cols 48..63, only 48,49 real
    float* __restrict__ spw = &Spk[wave][0];

    const float* npt = noise + baseFlat;
    float*       opt = out   + baseFlat;

    // ---- double-buffered noise prefetch (hides HBM latency across steps) ----
    float nz[2][4][8];
#pragma unroll
    for (int q = 0; q < 4; ++q) {
        const bool valid = (q < 3) || valid3;
#pragma unroll
        for (int r = 0; r < 8; ++r)
            nz[0][q][r] = valid ? npt[r * ROW_STRIDE + q * 16] : 0.0f;
    }

#pragma unroll 2
    for (int t = 0; t < TT; ++t) {
        const int cur = t & 1;

        // issue NEXT step's noise loads first; they overlap this step's matmul
        if (t + 1 < TT) {
            const float* npn = npt + STEP_STRIDE;
#pragma unroll
            for (int q = 0; q < 4; ++q) {
                const bool valid = (q < 3) || valid3;
#pragma unroll
                for (int r = 0; r < 8; ++r)
                    nz[cur ^ 1][q][r] = valid ? npn[r * ROW_STRIDE + q * 16] : 0.0f;
            }
        }

        // ---- reverb = spikes(16xKP) @ Wm(KPxNPAD) via f32 WMMA 16x16x4 ----
        v8f acc[4] = {};
#pragma unroll
        for (int k0 = 0; k0 < KP; k0 += 4) {
            // A fragment: M = nlo, K = k0 + koff + {0,1}  (contiguous -> b64)
            v2f a = *(const v2f*)&spw[nlo * KP + k0 + koff];
#pragma unroll
            for (int q = 0; q < 4; ++q) {
                // B fragment: K = k0 + koff + {0,1}, N = q*16 + nlo (contiguous in WldsT)
                v2f b = *(const v2f*)&WldsT[(q * 16 + nlo) * KP + k0 + koff];
                acc[q] = __builtin_amdgcn_wmma_f32_16x16x4_f32(
                    false, a, false, b, (short)0, acc[q], false, false);
            }
        }

        // ---- LIF update, threshold, reset; write spikes out + to LDS ----
#pragma unroll
        for (int q = 0; q < 4; ++q) {
            const bool valid = (q < 3) || valid3;
#pragma unroll
            for (int r = 0; r < 8; ++r) {
                float v = vst[q][r] * alpha + acc[q][r] + nz[cur][q][r] * 0.02f;
                if (t == 0 && valid)
                    v += pattern[baseFlat + r * ROW_STRIDE + q * 16];
                float sp = (v >= 1.0f) ? 1.0f : 0.0f;
                vst[q][r] = (sp != 0.0f) ? 0.0f : v;
                if (valid) {
                    opt[r * ROW_STRIDE + q * 16] = sp;
                    // next step's A operand: [M = r+8*hi][K = col]
                    spw[(r + 8 * hi) * KP + q * 16 + nlo] = sp;
                }
            }
        }

        npt += STEP_STRIDE;
        opt += STEP_STRIDE;
    }
}

extern "C" void kernel_launch(void* const* d_in, const int* in_sizes, int n_in,
                              void* d_out, int out_size, void* d_ws, size_t ws_size,
                              hipStream_t stream) {
    const float* pattern = (const float*)d_in[0];  // (2048, 7, 50)
    const float* noise   = (const float*)d_in[1];  // (128, 2048, 7, 50)
    const float* W       = (const float*)d_in[2];  // (7, 50, 50)
    float* out = (float*)d_out;                    // (128, 2048, 350)

    dim3 grid(NSLOT * (NB / 64));  // 224 blocks
    wm_lif_kernel<<<grid, 128, 0, stream>>>(pattern, noise, W, out);
}